// SimpleGCN_31602369364481
// MI455X (gfx1250) — compile-verified
//
#include <hip/hip_runtime.h>
#include <hip/hip_bf16.h>
#include <math.h>

typedef __attribute__((ext_vector_type(2))) float v2f;
typedef __attribute__((ext_vector_type(4))) float v4f;
typedef __attribute__((ext_vector_type(8))) float v8f;

#define LD 32  // padded row stride (floats) for node-feature buffers -> 128B rows

// ---------------- utility ----------------

__global__ void k_zero4(float4* p, int n4) {
  int i = blockIdx.x * blockDim.x + threadIdx.x;
  if (i < n4) p[i] = make_float4(0.f, 0.f, 0.f, 0.f);
}

__global__ void k_pack_x(const float* __restrict__ x, float* __restrict__ S, int F, int n) {
  int i = blockIdx.x * blockDim.x + threadIdx.x;
  if (i >= n) return;
  const float* xr = x + (size_t)i * F;
  float* sr = S + (size_t)i * LD;
  for (int f = 0; f < F; ++f) sr[f] = xr[f];
}

__global__ void k_tanh_rows(float* __restrict__ B, int Fout, int n) {
  int i = blockIdx.x * blockDim.x + threadIdx.x;
  if (i >= n) return;
  float* r = B + (size_t)i * LD;
  for (int f = 0; f < Fout; ++f) r[f] = tanhf(r[f]);
}

// ---------------- degree / normalization ----------------

__global__ void k_fill_deg(float* deg, int n) {
  int i = blockIdx.x * blockDim.x + threadIdx.x;
  if (i < n) deg[i] = 1.0f;  // self-loop contributes 1 to every node's degree
}

__global__ void k_deg_accum(const int* __restrict__ dst, float* deg, int E) {
  int e = blockIdx.x * blockDim.x + threadIdx.x;
  if (e < E) atomicAdd(&deg[dst[e]], 1.0f);
}

__global__ void k_dinv(float* deg, int n) {
  int i = blockIdx.x * blockDim.x + threadIdx.x;
  if (i < n) {
    float d = deg[i];
    deg[i] = (d > 0.0f) ? rsqrtf(d) : 0.0f;  // in-place: deg -> dinv
  }
}

__global__ void k_norm(const int* __restrict__ src, const int* __restrict__ dst,
                       const float* __restrict__ dinv, float* __restrict__ nrm, int E) {
  int e = blockIdx.x * blockDim.x + threadIdx.x;
  if (e < E) nrm[e] = dinv[src[e]] * dinv[dst[e]];
}

// ---------------- dense GEMM via V_WMMA_F32_16X16X4_F32 ----------------
// C[n x Fout] = A[Npad x LD] @ W[Fin x Fout] (+ bias)
// A rows zero-padded -> unconditional aligned v2f (b64) loads.
// W staged zero-padded into LDS [32x32] -> unconditional ds loads.
// One wave per 16-row tile, 8 waves/block.
// Compile-time KS (K-steps of 4) and NT (1 or 2 column tiles) -> fully
// unrolled loop, no per-iteration control flow.

template <int KS, int NT>
__global__ void k_gemm_wmma_t(const float* __restrict__ A,
                              const float* __restrict__ W, const float* __restrict__ bias,
                              int Fin, int Fout, float* __restrict__ C, int n) {
  __shared__ float Wl[32 * 32];
  for (int idx = threadIdx.x; idx < 32 * 32; idx += 256) {
    const int k = idx >> 5, c = idx & 31;
    Wl[idx] = (k < Fin && c < Fout) ? W[k * Fout + c] : 0.0f;
  }
  __syncthreads();

  const int lane = threadIdx.x & 31;
  const int hi   = lane >> 4;
  const int lr   = lane & 15;
  const int row0 = (blockIdx.x * 8 + (threadIdx.x >> 5)) * 16;
  const int arow = row0 + lr;  // always < Npad (grid sized to Npad)

  v8f acc0 = {};
  v8f acc1 = {};
  const float* Ar = A + (size_t)arow * LD + 2 * hi;

#pragma unroll
  for (int ks = 0; ks < KS; ++ks) {
    const int ka = 4 * ks + 2 * hi;
    const v2f a = *(const v2f*)(Ar + 4 * ks);   // aligned b64, no predication
    v2f b0; b0.x = Wl[ka * 32 + lr]; b0.y = Wl[(ka + 1) * 32 + lr];
    acc0 = __builtin_amdgcn_wmma_f32_16x16x4_f32(false, a, false, b0,
                                                 (short)0, acc0, false, false);
    if (NT == 2) {
      v2f b1; b1.x = Wl[ka * 32 + 16 + lr]; b1.y = Wl[(ka + 1) * 32 + 16 + lr];
      acc1 = __builtin_amdgcn_wmma_f32_16x16x4_f32(false, a, false, b1,
                                                   (short)0, acc1, false, false);
    }
  }

  const float bv0 = (bias && lr < Fout) ? bias[lr] : 0.0f;
  const float bv1 = (bias && NT == 2 && 16 + lr < Fout) ? bias[16 + lr] : 0.0f;
#pragma unroll
  for (int r = 0; r < 8; ++r) {
    const int orow = row0 + r + 8 * hi;
    if (orow < n) {
      if (lr < Fout)                 C[(size_t)orow * LD + lr]      = acc0[r] + bv0;
      if (NT == 2 && 16 + lr < Fout) C[(size_t)orow * LD + 16 + lr] = acc1[r] + bv1;
    }
  }
}

// Generic fallback (runtime trip counts)
__global__ void k_gemm_wmma_gen(const float* __restrict__ A,
                                const float* __restrict__ W, const float* __restrict__ bias,
                                int Fin, int Fout, float* __restrict__ C, int n) {
  __shared__ float Wl[32 * 32];
  for (int idx = threadIdx.x; idx < 32 * 32; idx += 256) {
    const int k = idx >> 5, c = idx & 31;
    Wl[idx] = (k < Fin && c < Fout) ? W[k * Fout + c] : 0.0f;
  }
  __syncthreads();
  const int lane = threadIdx.x & 31;
  const int hi = lane >> 4, lr = lane & 15;
  const int row0 = (blockIdx.x * 8 + (threadIdx.x >> 5)) * 16;
  const int arow = row0 + lr;
  const bool nt2 = (Fout > 16);
  v8f acc0 = {}, acc1 = {};
  const float* Ar = A + (size_t)arow * LD + 2 * hi;
  const int ksteps = (Fin + 3) >> 2;
#pragma unroll 2
  for (int ks = 0; ks < ksteps; ++ks) {
    const int ka = 4 * ks + 2 * hi;
    const v2f a = *(const v2f*)(Ar + 4 * ks);
    v2f b0; b0.x = Wl[ka * 32 + lr]; b0.y = Wl[(ka + 1) * 32 + lr];
    acc0 = __builtin_amdgcn_wmma_f32_16x16x4_f32(false, a, false, b0,
                                                 (short)0, acc0, false, false);
    if (nt2) {
      v2f b1; b1.x = Wl[ka * 32 + 16 + lr]; b1.y = Wl[(ka + 1) * 32 + 16 + lr];
      acc1 = __builtin_amdgcn_wmma_f32_16x16x4_f32(false, a, false, b1,
                                                   (short)0, acc1, false, false);
    }
  }
  const float bv0 = (bias && lr < Fout) ? bias[lr] : 0.0f;
  const float bv1 = (bias && nt2 && 16 + lr < Fout) ? bias[16 + lr] : 0.0f;
#pragma unroll
  for (int r = 0; r < 8; ++r) {
    const int orow = row0 + r + 8 * hi;
    if (orow < n) {
      if (lr < Fout)             C[(size_t)orow * LD + lr]      = acc0[r] + bv0;
      if (nt2 && 16 + lr < Fout) C[(size_t)orow * LD + 16 + lr] = acc1[r] + bv1;
    }
  }
}

// ---------------- aggregation: self-loop init + edge scatter ----------------

__global__ void k_agg_init(const float* __restrict__ G, const float* __restrict__ dinv,
                           const float* __restrict__ bias, float* __restrict__ S,
                           int Fout, int n) {
  int i = blockIdx.x * blockDim.x + threadIdx.x;
  if (i >= n) return;
  const float w = dinv[i] * dinv[i];  // self-loop norm
  const float* g = G + (size_t)i * LD;
  float* s = S + (size_t)i * LD;
  for (int f = 0; f < Fout; ++f) s[f] = bias[f] + g[f] * w;
}

// Compile-time FOUT: float4 gathers + unrolled atomic burst (L2-resident).
template <int FOUT>
__global__ void k_scatter_t(const int* __restrict__ src, const int* __restrict__ dst,
                            const float* __restrict__ nrm, const float* __restrict__ G,
                            float* __restrict__ S, int E) {
  int e = blockIdx.x * blockDim.x + threadIdx.x;
  if (e >= E) return;
  const int s = src[e];
  const int d = dst[e];
  const float w = nrm[e];
  const float* g = G + (size_t)s * LD;
  float* o = S + (size_t)d * LD;
  constexpr int Q = FOUT / 4;
  const float4* g4 = (const float4*)g;
#pragma unroll
  for (int q = 0; q < Q; ++q) {
    const float4 v = g4[q];
    atomicAdd(&o[4 * q + 0], v.x * w);
    atomicAdd(&o[4 * q + 1], v.y * w);
    atomicAdd(&o[4 * q + 2], v.z * w);
    atomicAdd(&o[4 * q + 3], v.w * w);
  }
#pragma unroll
  for (int f = 4 * Q; f < FOUT; ++f) atomicAdd(&o[f], g[f] * w);
}

__global__ void k_scatter_gen(const int* __restrict__ src, const int* __restrict__ dst,
                              const float* __restrict__ nrm, const float* __restrict__ G,
                              float* __restrict__ S, int Fout, int E) {
  int e = blockIdx.x * blockDim.x + threadIdx.x;
  if (e >= E) return;
  const int s = src[e];
  const int d = dst[e];
  const float w = nrm[e];
  const float* g = G + (size_t)s * LD;
  float* o = S + (size_t)d * LD;
  for (int f = 0; f < Fout; ++f) atomicAdd(&o[f], g[f] * w);
}

// ---------------- edge classifier + e-buffer emission ----------------
// Fast path: compile-time M (multiple of 4) and CC. Nontemporal stores keep
// the ~350MB output stream from evicting the L2-resident graph buffers.
template <int M, int CC>
__global__ void k_edge_fast(const int* __restrict__ src, const int* __restrict__ dst,
                            const float* __restrict__ H, const float* __restrict__ Cw,
                            const float* __restrict__ Cb,
                            float* __restrict__ out, float* __restrict__ ebuf, int E) {
  int e = blockIdx.x * blockDim.x + threadIdx.x;
  if (e >= E) return;
  const int s = src[e];
  const int d = dst[e];
  float h2[2 * M];
  const float4* hs4 = (const float4*)(H + (size_t)s * LD);
  const float4* hd4 = (const float4*)(H + (size_t)d * LD);
#pragma unroll
  for (int q = 0; q < M / 4; ++q) {
    const float4 v = hs4[q];
    h2[4 * q + 0] = v.x; h2[4 * q + 1] = v.y; h2[4 * q + 2] = v.z; h2[4 * q + 3] = v.w;
  }
#pragma unroll
  for (int q = 0; q < M / 4; ++q) {
    const float4 v = hd4[q];
    h2[M + 4 * q + 0] = v.x; h2[M + 4 * q + 1] = v.y;
    h2[M + 4 * q + 2] = v.z; h2[M + 4 * q + 3] = v.w;
  }
  float acc[CC];
#pragma unroll
  for (int c = 0; c < CC; ++c) acc[c] = Cb[c];
#pragma unroll
  for (int j = 0; j < 2 * M; ++j) {
#pragma unroll
    for (int c = 0; c < CC; ++c) acc[c] += h2[j] * Cw[j * CC + c];  // uniform addr
  }
  v4f* er4 = (v4f*)(ebuf + (size_t)e * (2 * M));  // 2M*4B row: 16B aligned
#pragma unroll
  for (int q = 0; q < (2 * M) / 4; ++q) {
    v4f t;
    t.x = h2[4 * q + 0]; t.y = h2[4 * q + 1]; t.z = h2[4 * q + 2]; t.w = h2[4 * q + 3];
    __builtin_nontemporal_store(t, er4 + q);
  }
  float* orow = out + (size_t)e * CC;
#pragma unroll
  for (int c = 0; c < CC; ++c) __builtin_nontemporal_store(acc[c], orow + c);
}

// Generic fallback (runtime M, CC)
__global__ void k_edge_gen(const int* __restrict__ src, const int* __restrict__ dst,
                           const float* __restrict__ H, const float* __restrict__ Cw,
                           const float* __restrict__ Cb, int M, int Cc,
                           float* __restrict__ out, float* __restrict__ ebuf, int E) {
  int e = blockIdx.x * blockDim.x + threadIdx.x;
  if (e >= E) return;
  const int s = src[e];
  const int d = dst[e];
  const float* hs = H + (size_t)s * LD;
  const float* hd = H + (size_t)d * LD;
  float acc[8];
  for (int c = 0; c < Cc; ++c) acc[c] = Cb[c];
  float* er = ebuf + (size_t)e * (2 * M);
  for (int j = 0; j < M; ++j) {
    const float v = hs[j];
    er[j] = v;
    for (int c = 0; c < Cc; ++c) acc[c] += v * Cw[j * Cc + c];
  }
  for (int j = 0; j < M; ++j) {
    const float v = hd[j];
    er[M + j] = v;
    for (int c = 0; c < Cc; ++c) acc[c] += v * Cw[(M + j) * Cc + c];
  }
  float* orow = out + (size_t)e * Cc;
  for (int c = 0; c < Cc; ++c) orow[c] = acc[c];
}

// ---------------- host-side dispatch helpers ----------------

static void launch_gemm(const float* A, const float* W, const float* bias,
                        int Fin, int Fout, float* C, int n, int gB, hipStream_t stream) {
  const int ks = (Fin + 3) >> 2;
  const int nt = (Fout > 16) ? 2 : 1;
  const int tag = ks * 2 + (nt - 1);
#define GEMM_CASE(KSV, NTV)                                                      \
  case (KSV * 2 + (NTV - 1)):                                                    \
    k_gemm_wmma_t<KSV, NTV><<<gB, 256, 0, stream>>>(A, W, bias, Fin, Fout, C, n);\
    break;
  switch (tag) {
    GEMM_CASE(1, 1) GEMM_CASE(1, 2)
    GEMM_CASE(2, 1) GEMM_CASE(2, 2)
    GEMM_CASE(3, 1) GEMM_CASE(3, 2)
    GEMM_CASE(4, 1) GEMM_CASE(4, 2)
    GEMM_CASE(5, 1) GEMM_CASE(5, 2)
    GEMM_CASE(6, 1) GEMM_CASE(6, 2)
    GEMM_CASE(7, 1) GEMM_CASE(7, 2)
    GEMM_CASE(8, 1) GEMM_CASE(8, 2)
    default:
      k_gemm_wmma_gen<<<gB, 256, 0, stream>>>(A, W, bias, Fin, Fout, C, n);
      break;
  }
#undef GEMM_CASE
}

static void launch_scatter(const int* src, const int* dst, const float* nrm,
                           const float* G, float* S, int Fout, int E, int bE,
                           hipStream_t stream) {
  switch (Fout) {
    case 12: k_scatter_t<12><<<bE, 256, 0, stream>>>(src, dst, nrm, G, S, E); break;
    case 16: k_scatter_t<16><<<bE, 256, 0, stream>>>(src, dst, nrm, G, S, E); break;
    case 18: k_scatter_t<18><<<bE, 256, 0, stream>>>(src, dst, nrm, G, S, E); break;
    case 20: k_scatter_t<20><<<bE, 256, 0, stream>>>(src, dst, nrm, G, S, E); break;
    case 24: k_scatter_t<24><<<bE, 256, 0, stream>>>(src, dst, nrm, G, S, E); break;
    case 32: k_scatter_t<32><<<bE, 256, 0, stream>>>(src, dst, nrm, G, S, E); break;
    default: k_scatter_gen<<<bE, 256, 0, stream>>>(src, dst, nrm, G, S, Fout, E); break;
  }
}

// ---------------- host orchestration ----------------

extern "C" void kernel_launch(void* const* d_in, const int* in_sizes, int n_in,
                              void* d_out, int out_size, void* d_ws, size_t ws_size,
                              hipStream_t stream) {
  const float* x   = (const float*)d_in[0];
  const int*   ei  = (const int*)  d_in[1];
  // d_in[2] = batch (unused)
  const float* W1  = (const float*)d_in[3];  const float* b1  = (const float*)d_in[4];
  const float* W2  = (const float*)d_in[5];  const float* b2  = (const float*)d_in[6];
  const float* W3  = (const float*)d_in[7];  const float* b3  = (const float*)d_in[8];
  const float* L1w = (const float*)d_in[9];  const float* L1b = (const float*)d_in[10];
  const float* L2w = (const float*)d_in[11]; const float* L2b = (const float*)d_in[12];
  const float* L3w = (const float*)d_in[13]; const float* L3b = (const float*)d_in[14];
  const float* Cw  = (const float*)d_in[15]; const float* Cb  = (const float*)d_in[16];

  const int N  = in_sizes[2];
  const int F  = in_sizes[0] / N;
  const int E  = in_sizes[1] / 2;
  const int H1 = in_sizes[4];   // 18
  const int H2 = in_sizes[6];   // 24
  const int H3 = in_sizes[8];   // 2F = 20
  const int M1 = in_sizes[10];  // 24
  const int M2 = in_sizes[12];  // 18
  const int M3 = in_sizes[14];  // 12
  const int Cc = in_sizes[16];  // 3

  const int* src = ei;
  const int* dst = ei + E;

  const int Npad = ((N + 127) / 128) * 128;

  // workspace layout (floats): deg/dinv [N] | norm [E] | G [Npad*LD] | S [Npad*LD]
  float* deg = (float*)d_ws;
  float* nrm = deg + N;
  float* G   = nrm + E;
  float* S   = G + (size_t)Npad * LD;

  const int T = 256;
  const int bN = (N + T - 1) / T;
  const int bE = (E + T - 1) / T;
  const int gB = Npad / 128;           // 8 waves/block, 16 rows/wave
  const int z4 = (Npad * LD) / 4;      // float4 count per buffer
  const int bZ = (z4 + T - 1) / T;

  // zero padded buffers (pad rows/cols must be 0 so GEMM loads are unconditional)
  k_zero4<<<bZ, T, 0, stream>>>((float4*)G, z4);
  k_zero4<<<bZ, T, 0, stream>>>((float4*)S, z4);

  // normalization: deg = 1 + count(dst); dinv = rsqrt(deg); norm[e] = dinv[s]*dinv[d]
  k_fill_deg <<<bN, T, 0, stream>>>(deg, N);
  k_deg_accum<<<bE, T, 0, stream>>>(dst, deg, E);
  k_dinv     <<<bN, T, 0, stream>>>(deg, N);
  k_norm     <<<bE, T, 0, stream>>>(src, dst, deg, nrm, E);

  // pack x into padded layout (S), then conv1: G = S @ W1
  k_pack_x<<<bN, T, 0, stream>>>(x, S, F, N);
  launch_gemm(S, W1, nullptr, F, H1, G, N, gB, stream);
  k_agg_init<<<bN, T, 0, stream>>>(G, deg, b1, S, H1, N);
  launch_scatter(src, dst, nrm, G, S, H1, E, bE, stream);
  k_tanh_rows<<<bN, T, 0, stream>>>(S, H1, N);

  // conv2
  launch_gemm(S, W2, nullptr, H1, H2, G, N, gB, stream);
  k_agg_init<<<bN, T, 0, stream>>>(G, deg, b2, S, H2, N);
  launch_scatter(src, dst, nrm, G, S, H2, E, bE, stream);
  k_tanh_rows<<<bN, T, 0, stream>>>(S, H2, N);

  // conv3
  launch_gemm(S, W3, nullptr, H2, H3, G, N, gB, stream);
  k_agg_init<<<bN, T, 0, stream>>>(G, deg, b3, S, H3, N);
  launch_scatter(src, dst, nrm, G, S, H3, E, bE, stream);
  k_tanh_rows<<<bN, T, 0, stream>>>(S, H3, N);

  // MLP: h = tanh(S @ L1w + L1b); tanh(@L2w+L2b); tanh(@L3w+L3b)
  launch_gemm(S, L1w, L1b, H3, M1, G, N, gB, stream);
  k_tanh_rows<<<bN, T, 0, stream>>>(G, M1, N);
  launch_gemm(G, L2w, L2b, M1, M2, S, N, gB, stream);
  k_tanh_rows<<<bN, T, 0, stream>>>(S, M2, N);
  launch_gemm(S, L3w, L3b, M2, M3, G, N, gB, stream);
  k_tanh_rows<<<bN, T, 0, stream>>>(G, M3, N);

  // edge classifier + e emission; d_out = [out (E*Cc) | e (E*2*M3)]
  float* outp = (float*)d_out;
  float* ebuf = outp + (size_t)E * Cc;
  if (M3 == 12 && Cc == 3) {
    k_edge_fast<12, 3><<<bE, T, 0, stream>>>(src, dst, G, Cw, Cb, outp, ebuf, E);
  } else {
    k_edge_gen<<<bE, T, 0, stream>>>(src, dst, G, Cw, Cb, M3, Cc, outp, ebuf, E);
  }
}